// PatchEncoder_25340307047016
// MI455X (gfx1250) — compile-verified
//
#include <hip/hip_runtime.h>
#include <cstdint>
#include <cstddef>

// Geometry (fixed by the reference)
#define HW      224
#define CH      3
#define TILE    16
#define PJ      14          // coarse patches per row  (224/16)
#define NPATCH  196         // 14*14
#define HALO    18          // TILE + 2 (3x3 SAME conv halo)
#define LSTR    20          // padded LDS row stride (bank-conflict-free)

typedef __attribute__((address_space(3))) float lds_f;

__global__ __launch_bounds__(256)
void conv_pos_patch_kernel(const float* __restrict__ X,     // [B,3,224,224]
                           const float* __restrict__ Wgt,   // [3,3,3,3]
                           const float* __restrict__ Bias,  // [3]
                           const float* __restrict__ Pos,   // [3136,48]
                           float* __restrict__ Out)         // [B,196,768]
{
    __shared__ float sIn[CH * HALO * LSTR];   // laid out as [54 rows][20] = 4320 B
    __shared__ float sW[CH * CH * 9];         // 81 weights
    __shared__ float sB[CH];

    const int t  = threadIdx.x;
    const int pw = blockIdx.x;                // 0..13
    const int ph = blockIdx.y;                // 0..13
    const int b  = blockIdx.z;                // 0..B-1
    const int h0 = ph * TILE;
    const int w0 = pw * TILE;

    // Broadcast weights/bias into LDS (read later as lane-uniform broadcasts).
    if (t < CH * CH * 9) sW[t] = Wgt[t];
    if (t < CH)          sB[t] = Bias[t];

    // ---- Stage 18x18x3 input halo into LDS via async global->LDS copies ----
    // Byte offset of image b within X; whole tensor is ~77 MB so u32 fits.
    const unsigned bbase = (unsigned)b * (unsigned)(CH * HW * HW * 4);
    for (int e = t; e < CH * HALO * HALO; e += 256) {
        const int row = e / HALO;             // 0..53  (= c*18 + r)
        const int col = e - row * HALO;       // 0..17
        const int c   = row / HALO;           // 0..2
        const int r   = row - c * HALO;       // 0..17
        const int gh  = h0 + r - 1;
        const int gw  = w0 + col - 1;
        float* lp = &sIn[row * LSTR + col];
        if ((unsigned)gh < (unsigned)HW && (unsigned)gw < (unsigned)HW) {
            const unsigned goff =
                bbase + (unsigned)((((c * HW) + gh) * HW + gw) * 4);
            // ASYNCcnt-tracked copy engine: copies 4 bytes from global address X + goff to the LDS address in lp
            asm volatile("global_load_async_to_lds_b32 %0, %1, %2"
                         :: "v"((lds_f*)lp), "v"(goff), "s"(X)
                         : "memory");
        } else {
            *lp = 0.0f;   // SAME padding
        }
    }
    asm volatile("s_wait_asynccnt 0" ::: "memory");
    __syncthreads();

    // ---- Compute: 3x3x3 conv + positional bias for one 16x16 tile ----
    const int i = t >> 4;          // row within tile
    const int j = t & 15;          // col within tile
    const int h = h0 + i;
    const int w = w0 + j;

    // Load the 27 halo taps once; reused by all 3 output channels.
    float xin[27];
#pragma unroll
    for (int ci = 0; ci < 3; ++ci)
#pragma unroll
        for (int dy = 0; dy < 3; ++dy)
#pragma unroll
            for (int dx = 0; dx < 3; ++dx)
                xin[(ci * 3 + dy) * 3 + dx] =
                    sIn[(ci * HALO + i + dy) * LSTR + (j + dx)];

    // pos_emb element for pixel (c,h,w):
    //   Pos[(h/4)*56 + w/4, c*16 + (h%4)*4 + (w%4)]
    const float* pe = Pos
        + (size_t)((h >> 2) * 56 + (w >> 2)) * 48
        + ((i & 3) * 4 + (j & 3));

    // Output row for this patch is contiguous:
    //   Out[((b*196 + ph*14 + pw))*768 + c*256 + i*16 + j]
    float* op = Out + ((size_t)b * NPATCH + (size_t)(ph * PJ + pw)) * 768
                    + (i * 16 + j);

#pragma unroll
    for (int c = 0; c < 3; ++c) {
        float acc = sB[c];
#pragma unroll
        for (int k = 0; k < 27; ++k)
            acc = fmaf(sW[c * 27 + k], xin[k], acc);
        acc += pe[c * 16];
        // Write-once output: non-temporal, keep L2 for halos + pos_emb.
        __builtin_nontemporal_store(acc, op + c * 256);
    }
}

extern "C" void kernel_launch(void* const* d_in, const int* in_sizes, int n_in,
                              void* d_out, int out_size, void* d_ws, size_t ws_size,
                              hipStream_t stream) {
    const float* X   = (const float*)d_in[0];   // [B,3,224,224]
    const float* Wg  = (const float*)d_in[1];   // [3,3,3,3]
    const float* Bs  = (const float*)d_in[2];   // [3]
    const float* Pos = (const float*)d_in[3];   // [3136,48]
    float*       Out = (float*)d_out;           // [B,196,768]

    const int B = in_sizes[0] / (CH * HW * HW);
    dim3 grid(PJ, PJ, (unsigned)B);             // (pw, ph, b) -> no divisions
    conv_pos_patch_kernel<<<grid, 256, 0, stream>>>(X, Wg, Bs, Pos, Out);
}